// SCReGAT_27805618274981
// MI455X (gfx1250) — compile-verified
//
#include <hip/hip_runtime.h>
#include <math.h>

typedef float v2f __attribute__((ext_vector_type(2)));
typedef float v8f __attribute__((ext_vector_type(8)));

__device__ __forceinline__ float lrelu01(float v) { return v >= 0.f ? v : 0.01f * v; }
__device__ __forceinline__ unsigned f2ord(float f) {
  unsigned u = __float_as_uint(f);
  return (u & 0x80000000u) ? ~u : (u | 0x80000000u);
}
__device__ __forceinline__ float ord2f(unsigned u) {
  unsigned b = (u & 0x80000000u) ? (u ^ 0x80000000u) : ~u;
  return __uint_as_float(b);
}

// ---------------------------------------------------------------------------
// Precompute collapsed attention matrices and folded biases (tiny, 1 block).
//   Asrc1/Adst1 [8x16], Medge [8x16], Asrc2/Adst2 [16x16],
//   fb1/fb2 [16] = flat_b + gat_bias @ flat_w,  resS [32] = res_g1*rsqrt(1+eps)
// ---------------------------------------------------------------------------
__global__ void precompute_small(
    const float* __restrict__ gat1_w, const float* __restrict__ att_src1,
    const float* __restrict__ att_dst1, const float* __restrict__ lin_edge,
    const float* __restrict__ att_edge, const float* __restrict__ gat2_w,
    const float* __restrict__ att_src2, const float* __restrict__ att_dst2,
    const float* __restrict__ gat1_bias, const float* __restrict__ flat1_w,
    const float* __restrict__ flat1_b, const float* __restrict__ gat2_bias,
    const float* __restrict__ flat2_w, const float* __restrict__ flat2_b,
    const float* __restrict__ res_g1,
    float* __restrict__ Asrc1, float* __restrict__ Adst1, float* __restrict__ Medge,
    float* __restrict__ Asrc2, float* __restrict__ Adst2,
    float* __restrict__ fb1, float* __restrict__ fb2, float* __restrict__ resS) {
  int t = threadIdx.x;
  if (t < 128) {                       // [8][16] matrices for GAT1
    int i = t >> 4, h = t & 15;
    float a = 0.f, b = 0.f, m = 0.f;
    for (int c = 0; c < 16; ++c) {
      float w = gat1_w[i * 256 + h * 16 + c];
      a += w * att_src1[h * 16 + c];
      b += w * att_dst1[h * 16 + c];
      m += lin_edge[i * 256 + h * 16 + c] * att_edge[h * 16 + c];
    }
    Asrc1[t] = a; Adst1[t] = b; Medge[t] = m;
  } else if (t < 384) {                // [16][16] matrices for GAT2
    int u = t - 128; int i = u >> 4, h = u & 15;
    float a = 0.f, b = 0.f;
    for (int c = 0; c < 16; ++c) {
      float w = gat2_w[i * 256 + h * 16 + c];
      a += w * att_src2[h * 16 + c];
      b += w * att_dst2[h * 16 + c];
    }
    Asrc2[u] = a; Adst2[u] = b;
  } else if (t < 400) {                // folded flat biases
    int k = t - 384;
    float a = flat1_b[k], b = flat2_b[k];
    for (int j = 0; j < 256; ++j) {
      a += gat1_bias[j] * flat1_w[j * 16 + k];
      b += gat2_bias[j] * flat2_w[j * 16 + k];
    }
    fb1[k] = a; fb2[k] = b;
  } else if (t < 432) {                // BN scale for res layer 1
    int k = t - 400;
    resS[k] = res_g1[k] * rsqrtf(1.0f + 1e-5f);
  }
}

// ---------------------------------------------------------------------------
// Node MLP: 1 -> 128 -> 64 -> 8 (BN folded, LReLU 0.01); one node per thread.
// W2 (128x64 = 32KB) staged in LDS; broadcast reads -> no bank conflicts.
// ---------------------------------------------------------------------------
__global__ void node_mlp(const float* __restrict__ raw_x,
    const float* __restrict__ w1, const float* __restrict__ b1,
    const float* __restrict__ g1, const float* __restrict__ be1,
    const float* __restrict__ w2, const float* __restrict__ b2,
    const float* __restrict__ g2, const float* __restrict__ be2,
    const float* __restrict__ w3, const float* __restrict__ b3,
    float* __restrict__ data8, int N_) {
  __shared__ float sw2[128 * 64];
  for (int i = threadIdx.x; i < 128 * 64; i += blockDim.x) sw2[i] = w2[i];
  __syncthreads();
  int n = blockIdx.x * blockDim.x + threadIdx.x;
  if (n >= N_) return;
  const float rs = rsqrtf(1.0f + 1e-5f);
  float x = raw_x[n];
  float acc[64];
#pragma unroll
  for (int j = 0; j < 64; ++j) acc[j] = 0.f;
  for (int i = 0; i < 128; ++i) {
    float h = (x * w1[i] + b1[i]) * (g1[i] * rs) + be1[i];
    h = lrelu01(h);
    const float* wr = &sw2[i * 64];
#pragma unroll
    for (int j = 0; j < 64; ++j) acc[j] += h * wr[j];
  }
  float o[8];
#pragma unroll
  for (int k = 0; k < 8; ++k) o[k] = b3[k];
#pragma unroll
  for (int j = 0; j < 64; ++j) {
    float h = (acc[j] + b2[j]) * (g2[j] * rs) + be2[j];
    h = lrelu01(h);
#pragma unroll
    for (int k = 0; k < 8; ++k) o[k] += h * w3[j * 8 + k];
  }
#pragma unroll
  for (int k = 0; k < 8; ++k) data8[(size_t)n * 8 + k] = o[k];
}

// ---------------------------------------------------------------------------
// Edge MLP: [xi*xj, xi, xj] -> 32 -> 16 -> 8 (BN+LReLU each) -> tanh
// then collapse through Medge[8x16] directly to the edge attention partials.
// ---------------------------------------------------------------------------
__global__ void edge_mlp(const float* __restrict__ raw_x, const int* __restrict__ ei,
    int E_, const float* __restrict__ w1, const float* __restrict__ b1,
    const float* __restrict__ g1, const float* __restrict__ be1,
    const float* __restrict__ w2, const float* __restrict__ b2,
    const float* __restrict__ g2, const float* __restrict__ be2,
    const float* __restrict__ w3, const float* __restrict__ b3,
    const float* __restrict__ g3, const float* __restrict__ be3,
    const float* __restrict__ Medge, float* __restrict__ alphaOut) {
  int e = blockIdx.x * blockDim.x + threadIdx.x;
  if (e >= E_) return;
  const float rs = rsqrtf(1.0f + 1e-5f);
  float xi = raw_x[ei[e]];
  float xj = raw_x[ei[E_ + e]];
  float f0 = xi * xj, f1 = xi, f2 = xj;
  float h32[32];
#pragma unroll
  for (int j = 0; j < 32; ++j) {
    float v = f0 * w1[j] + f1 * w1[32 + j] + f2 * w1[64 + j] + b1[j];
    v = v * (g1[j] * rs) + be1[j];
    h32[j] = lrelu01(v);
  }
  float h16[16];
#pragma unroll
  for (int k = 0; k < 16; ++k) {
    float v = b2[k];
#pragma unroll
    for (int j = 0; j < 32; ++j) v += h32[j] * w2[j * 16 + k];
    v = v * (g2[k] * rs) + be2[k];
    h16[k] = lrelu01(v);
  }
  float h8[8];
#pragma unroll
  for (int i = 0; i < 8; ++i) {
    float v = b3[i];
#pragma unroll
    for (int k = 0; k < 16; ++k) v += h16[k] * w3[k * 8 + i];
    v = v * (g3[i] * rs) + be3[i];
    h8[i] = tanhf(lrelu01(v));
  }
#pragma unroll
  for (int hh = 0; hh < 16; ++hh) {
    float v = 0.f;
#pragma unroll
    for (int i = 0; i < 8; ++i) v += h8[i] * Medge[i * 16 + hh];
    alphaOut[(size_t)e * 16 + hh] = v;
  }
}

// ---------------------------------------------------------------------------
// Generic f32-exact WMMA matmul: Y = epilogue(X[nrows,K] @ W[K,ncols]).
// One wave per 16x16 tile, K stepped by 4 via V_WMMA_F32_16X16X4_F32.
// A layout: lanes 0-15 hold K={0,1}, lanes 16-31 hold K={2,3} (ISA 7.12.2).
// Epilogue: +bias[col]; optional *scale[col]+shift[col] (folded BN); leaky.
// ---------------------------------------------------------------------------
__global__ void wmma_mm(const float* __restrict__ X, const float* __restrict__ W,
                        float* __restrict__ Y, int nrows, int K, int ncols,
                        const float* __restrict__ bias,
                        const float* __restrict__ scale,
                        const float* __restrict__ shift, int act) {
  int wave = blockIdx.x * (blockDim.x >> 5) + (threadIdx.x >> 5);
  int rt_count = (nrows + 15) >> 4;
  int ct_count = ncols >> 4;
  if (wave >= rt_count * ct_count) return;   // wave-uniform: EXEC stays full
  int rt = wave % rt_count;
  int ct = wave / rt_count;
  int lane = threadIdx.x & 31;
  int half = lane >> 4;
  int l = lane & 15;
  int arow = rt * 16 + l; if (arow >= nrows) arow = nrows - 1;
  int col = ct * 16 + l;
  const float* Xr = X + (size_t)arow * K + half * 2;
  const float* Wc = W + col + (size_t)(half * 2) * ncols;
  v8f acc = {0.f, 0.f, 0.f, 0.f, 0.f, 0.f, 0.f, 0.f};
  for (int kb = 0; kb < K; kb += 4) {
    v2f a, b;
    a.x = Xr[kb];
    a.y = Xr[kb + 1];
    b.x = Wc[(size_t)kb * ncols];
    b.y = Wc[(size_t)(kb + 1) * ncols];
    acc = __builtin_amdgcn_wmma_f32_16x16x4_f32(false, a, false, b, (short)0, acc,
                                                false, false);
  }
  float bcol = bias ? bias[col] : 0.f;
  float sc = scale ? scale[col] : 1.f;
  float sh = shift ? shift[col] : 0.f;
#pragma unroll
  for (int v = 0; v < 8; ++v) {
    int row = rt * 16 + v + half * 8;
    if (row < nrows) {
      float y = acc[v] + bcol;
      y = y * sc + sh;
      if (act) y = lrelu01(y);
      Y[(size_t)row * ncols + col] = y;
    }
  }
}

// ---------------------------------------------------------------------------
// Segment softmax over destination nodes (3 passes), alpha updated in place.
// segmax is an ordered-uint encoding (memset 0 == -inf).
// ---------------------------------------------------------------------------
__global__ void gat_logits(float* __restrict__ alpha, const float* __restrict__ asrc,
                           const float* __restrict__ adst, const int* __restrict__ ei,
                           int E_, unsigned* __restrict__ segmax, int has_base) {
  int t = blockIdx.x * blockDim.x + threadIdx.x;
  if (t >= E_ * 16) return;
  int e = t >> 4, hh = t & 15;
  int s = ei[e], d = ei[E_ + e];
  float v = has_base ? alpha[t] : 0.f;
  v += asrc[(size_t)s * 16 + hh] + adst[(size_t)d * 16 + hh];
  v = v >= 0.f ? v : 0.2f * v;             // leaky 0.2 (GAT negative_slope)
  alpha[t] = v;
  atomicMax(&segmax[(size_t)d * 16 + hh], f2ord(v));
}

__global__ void gat_expsum(float* __restrict__ alpha, const int* __restrict__ ei,
                           int E_, const unsigned* __restrict__ segmax,
                           float* __restrict__ segsum) {
  int t = blockIdx.x * blockDim.x + threadIdx.x;
  if (t >= E_ * 16) return;
  int e = t >> 4, hh = t & 15;
  int d = ei[E_ + e];
  float v = expf(alpha[t] - ord2f(segmax[(size_t)d * 16 + hh]));
  alpha[t] = v;
  atomicAdd(&segsum[(size_t)d * 16 + hh], v);
}

__global__ void gat_norm(float* __restrict__ alpha, const int* __restrict__ ei,
                         int E_, const float* __restrict__ segsum) {
  int t = blockIdx.x * blockDim.x + threadIdx.x;
  if (t >= E_ * 16) return;
  int e = t >> 4, hh = t & 15;
  int d = ei[E_ + e];
  alpha[t] = alpha[t] / (segsum[(size_t)d * 16 + hh] + 1e-16f);
}

// One block (256 threads) per edge: coalesced 1KB gather + coalesced atomics.
__global__ void gat_aggregate(const float* __restrict__ h, const float* __restrict__ alpha,
                              const int* __restrict__ ei, int E_,
                              float* __restrict__ aggr) {
  int e = blockIdx.x;
  if (e >= E_) return;
  int t = threadIdx.x;                       // 0..255  (= h*16 + c)
  int s = ei[e], d = ei[E_ + e];
  float a = alpha[(size_t)e * 16 + (t >> 4)];
  atomicAdd(&aggr[(size_t)d * 256 + t], h[(size_t)s * 256 + t] * a);
}

__global__ void vec_add(const float* __restrict__ a, const float* __restrict__ b,
                        float* __restrict__ c, int n) {
  int i = blockIdx.x * blockDim.x + threadIdx.x;
  if (i < n) c[i] = a[i] + b[i];
}

// gene_out[i] = -log_softmax(r[i])[0] = max + log(sum exp(r-max)) - r[0]
__global__ void gene_out_k(const float* __restrict__ r, float* __restrict__ out, int G_) {
  int i = blockIdx.x * blockDim.x + threadIdx.x;
  if (i >= G_) return;
  const float* row = r + (size_t)i * 16;
  float m = row[0];
#pragma unroll
  for (int k = 1; k < 16; ++k) m = fmaxf(m, row[k]);
  float s = 0.f;
#pragma unroll
  for (int k = 0; k < 16; ++k) s += expf(row[k] - m);
  out[i] = m + logf(s) - row[0];
}

// ---------------------------------------------------------------------------
static inline void launch_mm(const float* X, const float* W, float* Y, int nrows,
                             int K, int ncols, const float* bias, const float* scale,
                             const float* shift, int act, hipStream_t s) {
  int tiles = ((nrows + 15) / 16) * (ncols / 16);
  int blocks = (tiles + 3) / 4;   // 4 waves (16x16 tiles) per 128-thread block
  wmma_mm<<<blocks, 128, 0, s>>>(X, W, Y, nrows, K, ncols, bias, scale, shift, act);
}

extern "C" void kernel_launch(void* const* d_in, const int* in_sizes, int n_in,
                              void* d_out, int out_size, void* d_ws, size_t ws_size,
                              hipStream_t stream) {
  (void)n_in; (void)ws_size;
  const float* raw_x = (const float*)d_in[1];
  const int* ei  = (const int*)d_in[2];
  const int* etf = (const int*)d_in[3];
  const int N  = in_sizes[1];
  const int E  = in_sizes[2] / 2;
  const int E2 = in_sizes[3] / 2;
  const int G  = out_size - E * 16;

#define PRM(i) ((const float*)d_in[(i)])
  // params flattened in setup_inputs insertion order, base index 7
  const float *node_w1 = PRM(7),  *node_b1 = PRM(8),  *node_g1 = PRM(9),  *node_be1 = PRM(10);
  const float *node_w2 = PRM(11), *node_b2 = PRM(12), *node_g2 = PRM(13), *node_be2 = PRM(14);
  const float *node_w3 = PRM(15), *node_b3 = PRM(16);
  const float *edge_w1 = PRM(17), *edge_b1 = PRM(18), *edge_g1 = PRM(19), *edge_be1 = PRM(20);
  const float *edge_w2 = PRM(21), *edge_b2 = PRM(22), *edge_g2 = PRM(23), *edge_be2 = PRM(24);
  const float *edge_w3 = PRM(25), *edge_b3 = PRM(26), *edge_g3 = PRM(27), *edge_be3 = PRM(28);
  const float *res_w1  = PRM(29), *res_b1  = PRM(30), *res_g1  = PRM(31), *res_be1  = PRM(32);
  const float *res_w2  = PRM(33), *res_b2  = PRM(34);
  const float *gat1_w = PRM(35), *gat1_as = PRM(36), *gat1_ad = PRM(37);
  const float *gat1_le = PRM(38), *gat1_ae = PRM(39), *gat1_b = PRM(40);
  const float *flat1_w = PRM(41), *flat1_b = PRM(42);
  const float *gat2_w = PRM(43), *gat2_as = PRM(44), *gat2_ad = PRM(45), *gat2_b = PRM(46);
  const float *flat2_w = PRM(47), *flat2_b = PRM(48);
#undef PRM

  // workspace layout (floats)
  float* wsf = (float*)d_ws;
  size_t off = 0;
  float* data8 = wsf + off; off += (size_t)N * 8;
  float* hbuf  = wsf + off; off += (size_t)N * 256;   // h (GAT1) then h2 (GAT2)
  float* aggr  = wsf + off; off += (size_t)N * 256;   // aggregation; later res hidden
  float* asrc  = wsf + off; off += (size_t)N * 16;    // later: d = d1+d2
  float* adst  = wsf + off; off += (size_t)N * 16;    // later: res output r
  unsigned* segmax = (unsigned*)(wsf + off); off += (size_t)N * 16;
  float* segsum = wsf + off; off += (size_t)N * 16;
  float* d1    = wsf + off; off += (size_t)N * 16;
  float* d2    = wsf + off; off += (size_t)N * 16;
  float* alpha2 = wsf + off; off += (size_t)E2 * 16;
  float* small = wsf + off;
  float *Asrc1 = small, *Adst1 = small + 128, *Medge = small + 256;
  float *Asrc2 = small + 384, *Adst2 = small + 640;
  float *fb1 = small + 896, *fb2 = small + 912, *resS = small + 928;

  float* out = (float*)d_out;
  float* alpha1 = out + G;

  // 0) collapsed attention matrices + folded biases
  precompute_small<<<1, 512, 0, stream>>>(gat1_w, gat1_as, gat1_ad, gat1_le, gat1_ae,
      gat2_w, gat2_as, gat2_ad, gat1_b, flat1_w, flat1_b, gat2_b, flat2_w, flat2_b,
      res_g1, Asrc1, Adst1, Medge, Asrc2, Adst2, fb1, fb2, resS);

  // 1) node + edge MLPs
  node_mlp<<<(N + 255) / 256, 256, 0, stream>>>(raw_x, node_w1, node_b1, node_g1,
      node_be1, node_w2, node_b2, node_g2, node_be2, node_w3, node_b3, data8, N);
  edge_mlp<<<(E + 255) / 256, 256, 0, stream>>>(raw_x, ei, E, edge_w1, edge_b1,
      edge_g1, edge_be1, edge_w2, edge_b2, edge_g2, edge_be2, edge_w3, edge_b3,
      edge_g3, edge_be3, Medge, alpha1);

  // 2) GAT1 projections (WMMA f32)
  launch_mm(data8, gat1_w, hbuf, N, 8, 256, nullptr, nullptr, nullptr, 0, stream);
  launch_mm(data8, Asrc1, asrc, N, 8, 16, nullptr, nullptr, nullptr, 0, stream);
  launch_mm(data8, Adst1, adst, N, 8, 16, nullptr, nullptr, nullptr, 0, stream);

  // 3) GAT1 segment softmax (alpha1 finalized in d_out)
  hipMemsetAsync(segmax, 0, (size_t)N * 16 * sizeof(unsigned), stream);
  hipMemsetAsync(segsum, 0, (size_t)N * 16 * sizeof(float), stream);
  int tE = E * 16, bE = (tE + 255) / 256;
  gat_logits<<<bE, 256, 0, stream>>>(alpha1, asrc, adst, ei, E, segmax, 1);
  gat_expsum<<<bE, 256, 0, stream>>>(alpha1, ei, E, segmax, segsum);
  gat_norm<<<bE, 256, 0, stream>>>(alpha1, ei, E, segsum);

  // 4) GAT1 aggregation + flat1
  hipMemsetAsync(aggr, 0, (size_t)N * 256 * sizeof(float), stream);
  gat_aggregate<<<E, 256, 0, stream>>>(hbuf, alpha1, ei, E, aggr);
  launch_mm(aggr, flat1_w, d1, N, 256, 16, fb1, nullptr, nullptr, 1, stream);

  // 5) GAT2 projections
  launch_mm(d1, gat2_w, hbuf, N, 16, 256, nullptr, nullptr, nullptr, 0, stream);
  launch_mm(d1, Asrc2, asrc, N, 16, 16, nullptr, nullptr, nullptr, 0, stream);
  launch_mm(d1, Adst2, adst, N, 16, 16, nullptr, nullptr, nullptr, 0, stream);

  // 6) GAT2 segment softmax (alpha2 in scratch)
  hipMemsetAsync(segmax, 0, (size_t)N * 16 * sizeof(unsigned), stream);
  hipMemsetAsync(segsum, 0, (size_t)N * 16 * sizeof(float), stream);
  int tE2 = E2 * 16, bE2 = (tE2 + 255) / 256;
  gat_logits<<<bE2, 256, 0, stream>>>(alpha2, asrc, adst, etf, E2, segmax, 0);
  gat_expsum<<<bE2, 256, 0, stream>>>(alpha2, etf, E2, segmax, segsum);
  gat_norm<<<bE2, 256, 0, stream>>>(alpha2, etf, E2, segsum);

  // 7) GAT2 aggregation + flat2
  hipMemsetAsync(aggr, 0, (size_t)N * 256 * sizeof(float), stream);
  gat_aggregate<<<E2, 256, 0, stream>>>(hbuf, alpha2, etf, E2, aggr);
  launch_mm(aggr, flat2_w, d2, N, 256, 16, fb2, nullptr, nullptr, 1, stream);

  // 8) residual MLP:  d = d1 + d2 ; r = lrelu(bn(d@w1+b1)) @ w2 + b2
  float* dres = asrc;                 // reuse (N*16)
  float* r32  = aggr;                 // reuse (N*32 fits in N*256)
  float* rout = adst;                 // reuse (N*16)
  vec_add<<<((N * 16) + 255) / 256, 256, 0, stream>>>(d1, d2, dres, N * 16);
  launch_mm(dres, res_w1, r32, N, 16, 32, res_b1, resS, res_be1, 1, stream);
  launch_mm(r32, res_w2, rout, N, 32, 16, res_b2, nullptr, nullptr, 0, stream);

  // 9) gene readout (gene nodes are the first G nodes per gene_id_vec)
  gene_out_k<<<(G + 255) / 256, 256, 0, stream>>>(rout, out, G);
}